// Free_Tokenizer_clip_54803782697341
// MI455X (gfx1250) — compile-verified
//
#include <hip/hip_runtime.h>
#include <stdint.h>

typedef float v2f __attribute__((ext_vector_type(2)));
typedef float v4f __attribute__((ext_vector_type(4)));
typedef float v8f __attribute__((ext_vector_type(8)));
typedef int   v4i __attribute__((vector_size(16)));   // matches builtin's vector_size pointee

typedef __attribute__((address_space(1))) v4i* as1_v4i_ptr;
typedef __attribute__((address_space(3))) v4i* as3_v4i_ptr;

#define Bsz   32768
#define Cch   4
#define Dd    64
#define Tt    16
#define Oo    32
#define ROWS  64      // batch rows per block
#define THREADS 128   // 4 wave32

#if defined(__gfx1250__) && \
    __has_builtin(__builtin_amdgcn_global_load_async_to_lds_b128) && \
    __has_builtin(__builtin_amdgcn_s_wait_asynccnt)
#define USE_ASYNC_LDS 1
#else
#define USE_ASYNC_LDS 0
#endif

// Factored algorithm (2x fewer matrix FLOPs than the reference einsum order):
//   Y[b,t,:]      = sum_c Wm[t,c] * x[b,c,:]               (fused into A-fragment build)
//   tokens[b,t,:] = relu( Y[b,t,:] @ Wg[t]^T + bg[t,:]*S[t] ),  S[t] = sum_c Wm[t,c]
__global__ __launch_bounds__(THREADS) void tok_kernel(
    const float* __restrict__ x,   // [B, C, D]
    const float* __restrict__ Wg,  // [T, O, D]
    const float* __restrict__ bg,  // [T, O]
    const float* __restrict__ Wm,  // [T, C]
    float* __restrict__ out)       // [B, T*O]
{
  // 64 KB: [row][c][k-chunk ^ (row&15)][k&3] -- straight 16B-chunk async copy of the x
  // tile; XOR swizzle on the 16B chunk index makes the ds_load_b64 reads conflict-free.
  __shared__ float xs[ROWS * 256];

  const int tid  = threadIdx.x;
  const int lane = tid & 31;
  const int wv   = tid >> 5;              // wave id 0..3
  const int b0   = blockIdx.x * ROWS;

  // Warm L2 for the weight stream (128 threads x 1KB stride covers all 128KB of Wg).
  __builtin_prefetch(Wg + (size_t)tid * 256, 0, 0);

  // ---- async-stage this wave's 16 rows (16KB) into LDS as 16B chunks ----
  {
    const int base_row = wv * 16;
#pragma unroll 8
    for (int i = 0; i < 32; ++i) {
      int task = lane + i * 32;
      int rl = task >> 6;           // row within wave tile (0..15)
      int c  = (task >> 4) & 3;     // channel
      int g  = task & 15;           // 16B k-chunk (4 floats)
      const float* gsrc = x + (size_t)(b0 + base_row + rl) * (Cch * Dd) + c * Dd + g * 4;
      float*       ldst = xs + (base_row + rl) * 256 + c * Dd + ((g ^ rl) << 2);
#if USE_ASYNC_LDS
      __builtin_amdgcn_global_load_async_to_lds_b128(
          (as1_v4i_ptr)(uintptr_t)gsrc,
          (as3_v4i_ptr)(uint32_t)(uintptr_t)ldst,
          0, 0);
#else
      *(v4f*)ldst = *(const v4f*)gsrc;
#endif
    }
#if USE_ASYNC_LDS
    __builtin_amdgcn_s_wait_asynccnt(0);
#endif
    __syncthreads();
  }

  // WMMA 16x16x4 f32 fragment geometry (wave32):
  //   A (16x4): lanes 0-15 row M=lane, K={k0,k0+1}; lanes 16-31 same M, K={k0+2,k0+3}
  //   B (4x16): lanes 0-15 col N=lane, K={k0,k0+1}; lanes 16-31 N=lane-16, K={k0+2,k0+3}
  //   C/D: VGPR i -> M = i (+8 for upper half), N = lane&15
  const int rlow = lane & 15;
  const int hi   = lane >> 4;                      // 0 = lower half (K+0,1), 1 = upper (K+2,3)
  const int rowl = wv * 16 + rlow;                 // A-matrix row this lane serves
  const float* lbase = xs + rowl * 256 + hi * 2;   // +2 floats: upper half K offset

  // ---- pull this lane's x slice into registers once (token-invariant) ----
  v2f xr0[16], xr1[16], xr2[16], xr3[16];          // xr{c}[s] = x[row][c][4s+2hi .. +1]
#pragma unroll
  for (int s = 0; s < 16; ++s) {
    const float* p = lbase + ((s ^ rlow) << 2);
    xr0[s] = *(const v2f*)(p + 0 * Dd);
    xr1[s] = *(const v2f*)(p + 1 * Dd);
    xr2[s] = *(const v2f*)(p + 2 * Dd);
    xr3[s] = *(const v2f*)(p + 3 * Dd);
  }

  for (int t = 0; t < Tt; ++t) {
    // merge weights (uniform -> SGPRs), splatted for packed FMA
    float wm0 = Wm[t * 4 + 0], wm1 = Wm[t * 4 + 1];
    float wm2 = Wm[t * 4 + 2], wm3 = Wm[t * 4 + 3];
    float St  = wm0 + wm1 + wm2 + wm3;
    v2f w0 = {wm0, wm0}, w1 = {wm1, wm1}, w2 = {wm2, wm2}, w3 = {wm3, wm3};

    // ---- batched preload of all 32 B fragments + biases (one clause, one wait) ----
    const float* wrow0 = Wg + (size_t)(t * Oo + rlow) * Dd + hi * 2;   // n0=0
    const float* wrow1 = wrow0 + 16 * Dd;                              // n0=16
    v2f bf0[16], bf1[16];
#pragma unroll
    for (int s = 0; s < 16; ++s) {
      bf0[s] = *(const v2f*)(wrow0 + s * 4);       // Wg[t][n][k], k = s*4 + 2*hi (+1)
      bf1[s] = *(const v2f*)(wrow1 + s * 4);
    }
    float bias0 = bg[t * Oo + rlow] * St;
    float bias1 = bg[t * Oo + 16 + rlow] * St;

    // ---- 32 WMMAs on two independent accumulator chains ----
    v8f acc0 = {};   // N tile 0..15
    v8f acc1 = {};   // N tile 16..31
#pragma unroll
    for (int s = 0; s < 16; ++s) {
      // fused channel merge: 1 pk_mul + 3 pk_fma (fp32 exact)
      v2f a = xr0[s] * w0 + xr1[s] * w1 + xr2[s] * w2 + xr3[s] * w3;
      acc0 = __builtin_amdgcn_wmma_f32_16x16x4_f32(false, a, false, bf0[s],
                                                   (short)0, acc0, false, false);
      acc1 = __builtin_amdgcn_wmma_f32_16x16x4_f32(false, a, false, bf1[s],
                                                   (short)0, acc1, false, false);
    }

    // bias*S[t] + relu + store; N depends only on lane&15 -> one bias scalar per N tile
    float* obase = out + (size_t)(b0 + wv * 16) * (Tt * Oo) + t * Oo + rlow;
#pragma unroll
    for (int i = 0; i < 8; ++i) {
      int m = i + hi * 8;                           // C/D layout M index
      float v0 = acc0[i] + bias0;
      float v1 = acc1[i] + bias1;
      obase[(size_t)m * (Tt * Oo) + 0 ] = v0 > 0.f ? v0 : 0.f;
      obase[(size_t)m * (Tt * Oo) + 16] = v1 > 0.f ? v1 : 0.f;
    }
  }
}

extern "C" void kernel_launch(void* const* d_in, const int* in_sizes, int n_in,
                              void* d_out, int out_size, void* d_ws, size_t ws_size,
                              hipStream_t stream) {
  const float* x  = (const float*)d_in[0];   // [B,C,D]
  const float* Wg = (const float*)d_in[1];   // [T,O,D]
  const float* bg = (const float*)d_in[2];   // [T,O]
  const float* Wm = (const float*)d_in[3];   // [T,C]
  float* out = (float*)d_out;                // [B, T*O]
  (void)in_sizes; (void)n_in; (void)out_size; (void)d_ws; (void)ws_size;

  dim3 grid(Bsz / ROWS);
  dim3 block(THREADS);
  tok_kernel<<<grid, block, 0, stream>>>(x, Wg, bg, Wm, out);
}